// HBEMD_F_67156108640795
// MI455X (gfx1250) — compile-verified
//
#include <hip/hip_runtime.h>

// Reference: out = rgb_to_hue(x) * W + b
//   x: (32, 3, 512, 512) fp32, W: scalar, b: scalar, out: (32, 1, 512, 512) fp32
//
// MI455X reasoning: 96 MB in + 32 MB out single-pass @ 23.3 TB/s => ~5.5 us
// floor; pure streaming elementwise (no contraction dim => WMMA inapplicable).
// Optimal kernel saturates HBM: clause-grouped global_load_b128 with TH=NT,
// NT b128 store, wave32-friendly 256-thread blocks, 65K waves in flight.
// v1 -> v2: replace the IEEE divide expansion (~10 VALU ops/pixel) with one
// v_rcp_f32 (TRANS op, co-executes with VALU) + v_fma_f32, dropping VALU
// intensity from ~2.0 to ~1.4 ops/byte so compute stays far under the roofline.

typedef __attribute__((ext_vector_type(4))) float v4f;

#define PLANE4   65536              // (512*512)/4 vec4s per channel plane
#define NIMG     32
#define N4_TOTAL (NIMG * PLANE4)    // 2,097,152 vec4 work items

__device__ __forceinline__ float hue_affine(float r, float g, float b,
                                            float w6, float bias) {
    float cmax  = fmaxf(r, fmaxf(g, b));
    float cmin  = fminf(r, fminf(g, b));
    float delta = cmax - cmin;

    // argmax with first-index tie-break (matches jnp.argmax over axis=1)
    bool rmax = (r >= g) && (r >= b);
    bool gmax = (g >= b);

    float num = rmax ? (g - b) : (gmax ? (b - r) : (r - g));
    float add = rmax ? 0.0f    : (gmax ? 2.0f    : 4.0f);

    float sd  = (delta == 0.0f) ? 1.0f : delta;   // safe_delta
    // |num| <= delta, so num*rcp(sd) is in [-1,1]; 1-ULP rcp is ample for hue.
    float inv = __builtin_amdgcn_rcpf(sd);        // v_rcp_f32 (TRANS)
    float v   = fmaf(num, inv, add);
    // jnp.mod(h0, 6): only the rmax branch can go negative (v in [-1,1]);
    // the others yield v >= 1, so an unconditional wrap is exact.
    v = (v < 0.0f) ? v + 6.0f : v;

    float h = (delta == 0.0f) ? 0.0f : v;         // delta==0 -> hue 0
    return fmaf(h, w6, bias);                     // (h/6)*W + b, W/6 prefolded
}

__global__ __launch_bounds__(256) void
hue_affine_kernel(const float* __restrict__ x,
                  const float* __restrict__ Wp,
                  const float* __restrict__ Bp,
                  float* __restrict__ out) {
    int t = blockIdx.x * blockDim.x + threadIdx.x;
    if (t >= N4_TOTAL) return;

    int  n    = t >> 16;                 // image index
    int  q    = t & (PLANE4 - 1);        // vec4 offset within a plane
    long base = (long)n * (3 * PLANE4) + q;

    const v4f* xp = (const v4f*)x;
    // Single-pass streaming: non-temporal loads (TH=NT) keep the 192MB L2
    // from retaining dead lines.
    v4f r = __builtin_nontemporal_load(xp + base);
    v4f g = __builtin_nontemporal_load(xp + base + PLANE4);
    v4f b = __builtin_nontemporal_load(xp + base + 2 * PLANE4);

    float w6   = Wp[0] * (1.0f / 6.0f);  // uniform -> scalar load + 1 smul
    float bias = Bp[0];

    v4f o;
    o.x = hue_affine(r.x, g.x, b.x, w6, bias);
    o.y = hue_affine(r.y, g.y, b.y, w6, bias);
    o.z = hue_affine(r.z, g.z, b.z, w6, bias);
    o.w = hue_affine(r.w, g.w, b.w, w6, bias);

    __builtin_nontemporal_store(o, (v4f*)out + t);
}

extern "C" void kernel_launch(void* const* d_in, const int* in_sizes, int n_in,
                              void* d_out, int out_size, void* d_ws, size_t ws_size,
                              hipStream_t stream) {
    const float* x   = (const float*)d_in[0];  // (32,3,512,512) fp32
    const float* W   = (const float*)d_in[1];  // 1 elem
    const float* B   = (const float*)d_in[2];  // 1 elem
    float*       out = (float*)d_out;          // (32,1,512,512) fp32

    const int block = 256;                     // 8 wave32 waves per block
    const int grid  = N4_TOTAL / block;        // 8192 blocks, exact cover
    hue_affine_kernel<<<grid, block, 0, stream>>>(x, W, B, out);
}